// PinUtilization_71519795413198
// MI455X (gfx1250) — compile-verified
//
#include <hip/hip_runtime.h>

// ---------------------------------------------------------------------------
// Pin-utilization density map (DREAMPlace-style) for MI455X / gfx1250.
// Atomic-bound scatter: 512x512 f32 bin grid (1 MB) is L2-resident; work is
// dominated by global_atomic_add_f32. Inputs streamed via CDNA5 async
// global->LDS copies (ASYNCcnt) with double buffering.
// ---------------------------------------------------------------------------

namespace {
constexpr int   kNumNodes    = 1200000;
constexpr int   kNumPhysical = 1000000;   // NUM_NODES - NUM_FILLER
constexpr int   kNB          = 512;       // bins per axis
constexpr float kXL          = 0.0f;
constexpr float kYL          = 0.0f;
constexpr float kBSX         = 1000.0f / 512.0f;   // 1.953125 (exact)
constexpr float kBSY         = 1000.0f / 512.0f;
constexpr float kInvBSX      = 512.0f / 1000.0f;
constexpr float kInvBSY      = 512.0f / 1000.0f;
constexpr float kStretch     = 1.414f;
constexpr float kOutScale    = 1.0f / (kBSX * kBSY * 0.5f);  // 1/(bx*by*cap)
constexpr int   KWIN         = 5;
constexpr int   BLOCK        = 256;
constexpr int   GRID         = 1024;
}

// Native L2-side float atomic add (agent scope, relaxed) -> global_atomic_add_f32
__device__ __forceinline__ void atomic_add_f32(float* p, float v) {
  __hip_atomic_fetch_add(p, v, __ATOMIC_RELAXED, __HIP_MEMORY_SCOPE_AGENT);
}

// --- Diagnostic probe (never launched): placed first so the disasm snippet
// --- shows how the float atomic lowers (expect a single global_atomic_add_f32).
__global__ void aaa_atomic_probe(float* __restrict__ p, float v) {
  atomic_add_f32(p + threadIdx.x, v);
}

// Per-node compute + guarded scatter. Zero-overlap contributions are skipped
// (exactly equivalent to the reference, which adds zeros at clamped indices).
__device__ __forceinline__ void scatter_node(float x, float y, float sx, float sy,
                                             float w, float* __restrict__ out) {
  const float hsx = 0.5f * fmaxf(sx, kBSX * kStretch);
  const float hsy = 0.5f * fmaxf(sy, kBSY * kStretch);
  const float cx = x + 0.5f * sx;
  const float cy = y + 0.5f * sy;
  const float xmin = cx - hsx, xmax = cx + hsx;
  const float ymin = cy - hsy, ymax = cy + hsy;
  // density * final output scale folded together
  const float dens = w * (0.25f * kOutScale) / (hsx * hsy);

  int bxl = (int)floorf((xmin - kXL) * kInvBSX);
  bxl = bxl < 0 ? 0 : (bxl > kNB - 1 ? kNB - 1 : bxl);
  int byl = (int)floorf((ymin - kYL) * kInvBSY);
  byl = byl < 0 ? 0 : (byl > kNB - 1 ? kNB - 1 : byl);

  float ovy[KWIN];
  int   iy[KWIN];
#pragma unroll
  for (int k = 0; k < KWIN; ++k) {
    const int   idx = byl + k;
    const float blo = kYL + (float)idx * kBSY;
    float ov = fminf(ymax, blo + kBSY) - fmaxf(ymin, blo);
    ov = (idx < kNB) ? fmaxf(ov, 0.0f) : 0.0f;
    ovy[k] = ov;
    iy[k]  = idx < kNB - 1 ? idx : kNB - 1;
  }

#pragma unroll
  for (int kx = 0; kx < KWIN; ++kx) {
    const int   idx = bxl + kx;
    const float blo = kXL + (float)idx * kBSX;
    float ov = fminf(xmax, blo + kBSX) - fmaxf(xmin, blo);
    ov = (idx < kNB) ? fmaxf(ov, 0.0f) : 0.0f;
    if (ov > 0.0f) {                       // idx < kNB guaranteed here
      const float a   = ov * dens;
      float*      row = out + (size_t)idx * kNB;
#pragma unroll
      for (int ky = 0; ky < KWIN; ++ky) {
        if (ovy[ky] > 0.0f) {
          atomic_add_f32(row + iy[ky], a * ovy[ky]);
        }
      }
    }
  }
}

#if defined(__HIP_DEVICE_COMPILE__) &&                                   \
    __has_builtin(__builtin_amdgcn_global_load_async_to_lds_b32) &&      \
    __has_builtin(__builtin_amdgcn_s_wait_asynccnt)
#define PIN_ASYNC_LDS 1
#else
#define PIN_ASYNC_LDS 0
#endif

#if PIN_ASYNC_LDS
#define AS_GLOBAL __attribute__((address_space(1)))
#define AS_LOCAL  __attribute__((address_space(3)))
__device__ __forceinline__ void async_ld_f32(const float* g, float* l) {
  __builtin_amdgcn_global_load_async_to_lds_b32(
      (AS_GLOBAL int*)(const_cast<float*>(g)), (AS_LOCAL int*)(l),
      /*offset=*/0, /*cpol=*/0);
}
#endif

__global__ void __launch_bounds__(BLOCK)
pin_util_scatter(const float* __restrict__ pos,
                 const float* __restrict__ nsx,
                 const float* __restrict__ nsy,
                 const float* __restrict__ pw,
                 float* __restrict__ out) {
  const int tid      = threadIdx.x;
  const int numTiles = (kNumPhysical + BLOCK - 1) / BLOCK;

#if PIN_ASYNC_LDS
  // Double-buffered async staging: tile t+GRID's 5 input streams are in
  // flight on the async copy engine while tile t issues its atomics.
  __shared__ float stage[2][5][BLOCK];

  auto issue = [&](int tile, int p) {
    int i = tile * BLOCK + tid;
    if (i >= kNumPhysical) i = kNumPhysical - 1;  // harmless duplicate fetch
    async_ld_f32(pos + i,             &stage[p][0][tid]);
    async_ld_f32(pos + kNumNodes + i, &stage[p][1][tid]);
    async_ld_f32(nsx + i,             &stage[p][2][tid]);
    async_ld_f32(nsy + i,             &stage[p][3][tid]);
    async_ld_f32(pw + i,              &stage[p][4][tid]);
  };

  int t = (int)blockIdx.x;
  if (t < numTiles) issue(t, 0);
  int p = 0;
  for (; t < numTiles; t += (int)gridDim.x) {
    const int tn = t + (int)gridDim.x;
    if (tn < numTiles) {
      issue(tn, p ^ 1);
      // Async loads complete in order: <=5 outstanding means the 5 loads of
      // the *current* buffer have landed in LDS; next buffer may be in flight.
      __builtin_amdgcn_s_wait_asynccnt(5);
    } else {
      __builtin_amdgcn_s_wait_asynccnt(0);
    }
    const float x  = stage[p][0][tid];
    const float y  = stage[p][1][tid];
    const float sx = stage[p][2][tid];
    const float sy = stage[p][3][tid];
    const float w  = stage[p][4][tid];
    const int node = t * BLOCK + tid;
    if (node < kNumPhysical) {
      scatter_node(x, y, sx, sy, w, out);
    }
    p ^= 1;
  }
#else
  // Fallback: direct global loads, grid-stride.
  for (int i = (int)blockIdx.x * BLOCK + tid; i < kNumPhysical;
       i += (int)gridDim.x * BLOCK) {
    scatter_node(pos[i], pos[kNumNodes + i], nsx[i], nsy[i], pw[i], out);
  }
#endif
}

__global__ void pin_util_zero(float* __restrict__ out, int n) {
  const int i = blockIdx.x * blockDim.x + threadIdx.x;
  if (i < n) out[i] = 0.0f;
}

extern "C" void kernel_launch(void* const* d_in, const int* in_sizes, int n_in,
                              void* d_out, int out_size, void* d_ws, size_t ws_size,
                              hipStream_t stream) {
  (void)in_sizes; (void)n_in; (void)d_ws; (void)ws_size;
  const float* pos = (const float*)d_in[0];
  const float* nsx = (const float*)d_in[1];
  const float* nsy = (const float*)d_in[2];
  const float* pw  = (const float*)d_in[3];
  float*       out = (float*)d_out;

  const int nb2 = kNB * kNB;  // 262144 == out_size
  (void)out_size;
  pin_util_zero<<<(nb2 + 255) / 256, 256, 0, stream>>>(out, nb2);
  pin_util_scatter<<<GRID, BLOCK, 0, stream>>>(pos, nsx, nsy, pw, out);
}